// VectorQuantizer_3642132267104
// MI455X (gfx1250) — compile-verified
//
#include <hip/hip_runtime.h>

typedef __attribute__((ext_vector_type(2))) float v2f;
typedef __attribute__((ext_vector_type(8))) float v8f;
typedef __attribute__((ext_vector_type(4))) int   v4i;

#define EMB_K 1024
#define DIM 256
#define M_TOTAL 16384          // 8 * 2048
#define ROWS_PER_WG 128        // 8 waves * 16 rows
#define CODES_PER_ITER 32      // two 16-code WMMA tiles per pass, shared A fragment
#define NITER (EMB_K / CODES_PER_ITER)   // 32
#define XPITCH 258             // 256 + 2 pad: bank offset 2/row, keeps 8B align
#define LDS_X_ELEMS (ROWS_PER_WG * XPITCH)
#define LDS_E_ELEMS (CODES_PER_ITER * XPITCH)
// Ping-pong e-tile buffers allocated unconditionally so host/device agree on size.
#define LDS_TOTAL_ELEMS (LDS_X_ELEMS + 2 * LDS_E_ELEMS + EMB_K)

#define HAVE_ASYNC_LDS (__has_builtin(__builtin_amdgcn_global_load_async_to_lds_b128) && \
                        __has_builtin(__builtin_amdgcn_s_wait_asynccnt))

#if HAVE_ASYNC_LDS
// Async global->LDS copy of 16B per lane (ASYNCcnt-tracked, writes LDS directly).
__device__ __forceinline__ void async_copy16(const float* g, float* l) {
    float* gg = const_cast<float*>(g);
    __builtin_amdgcn_global_load_async_to_lds_b128(
        (__attribute__((address_space(1))) v4i*)gg,
        (__attribute__((address_space(3))) v4i*)l, 0, 0);
}
#endif

// ---------------- kernel 0: e_sq[k] = sum_d emb[k][d]^2 ----------------
__global__ void vq_esq(const float* __restrict__ emb, float* __restrict__ esq) {
    int c = blockIdx.x * blockDim.x + threadIdx.x;
    if (c < EMB_K) {
        const float* row = emb + (size_t)c * DIM;
        float s = 0.f;
#pragma unroll 4
        for (int d = 0; d < DIM; d += 4) {
            float4 v = *(const float4*)(row + d);
            s += v.x * v.x + v.y * v.y + v.z * v.z + v.w * v.w;
        }
        esq[c] = s;
    }
}

// ---------------- kernel 1: fused distances + argmin via f32 WMMA ----------------
__global__ void __launch_bounds__(256) vq_argmin(const float* __restrict__ x,
                                                 const float* __restrict__ emb,
                                                 const float* __restrict__ esq,
                                                 int* __restrict__ idx_out) {
    extern __shared__ float lds[];
    float* lds_x  = lds;                               // [128][258]
    float* ebuf0  = lds + LDS_X_ELEMS;                 // [32][258]
    float* ebuf1  = ebuf0 + LDS_E_ELEMS;               // [32][258]
    float* lds_q  = ebuf1 + LDS_E_ELEMS;               // [1024] e_sq

    const int tid  = threadIdx.x;
    const int wave = tid >> 5;
    const int lane = tid & 31;
    const int half = lane >> 4;         // 0: lanes 0-15, 1: lanes 16-31
    const int l16  = lane & 15;
    const int rbase = blockIdx.x * ROWS_PER_WG;

#if HAVE_ASYNC_LDS
    // Issue async staging of code tile 0 (ASYNCcnt path), overlaps x staging below.
#pragma unroll
    for (int j = 0; j < 8; ++j) {
        int i  = tid + j * 256;
        int r  = i >> 6;
        int c4 = i & 63;
        async_copy16(emb + (size_t)r * DIM + c4 * 4,
                     ebuf0 + r * XPITCH + c4 * 4);
    }
#endif

    // Stage e_sq (1024 f32 = 4KB) once.
    for (int i = tid; i < EMB_K; i += 256) lds_q[i] = esq[i];

    // Cooperative, coalesced load of the 128x256 x-tile into LDS.
    for (int i = tid; i < ROWS_PER_WG * (DIM / 4); i += 256) {
        int r  = i >> 6;                // / (DIM/4)
        int c4 = i & 63;
        float4 v = *(const float4*)(x + (size_t)(rbase + r) * DIM + c4 * 4);
        float* dst = lds_x + r * XPITCH + c4 * 4;
        dst[0] = v.x; dst[1] = v.y; dst[2] = v.z; dst[3] = v.w;
    }

    // Lane-local running argmin over this lane's code column (codes == l16 mod 16).
    float best[8];
    int   btile[8];
#pragma unroll
    for (int r = 0; r < 8; ++r) { best[r] = 3.4e38f; btile[r] = 0; }

    // A fragment (16x4 f32): lane -> row = l16, K pair = 2*half (+0,+1)
    const float* aptr = lds_x + (wave * 16 + l16) * XPITCH + 2 * half;
    // B fragment lane offset (4x16 f32): lane -> code col = l16, K pair = 2*half
    const int boff = l16 * XPITCH + 2 * half;

#if !HAVE_ASYNC_LDS
    // Fallback: register double-buffer, 8 x float4 per thread.
    float4 pf[8];
#pragma unroll
    for (int j = 0; j < 8; ++j) {
        int i  = tid + j * 256;
        int r  = i >> 6;
        int c4 = i & 63;
        pf[j] = *(const float4*)(emb + (size_t)r * DIM + c4 * 4);
    }
#endif

    for (int it = 0; it < NITER; ++it) {
#if HAVE_ASYNC_LDS
        float* cur = (it & 1) ? ebuf1 : ebuf0;
        float* nxt = (it & 1) ? ebuf0 : ebuf1;
        if (it + 1 < NITER) {
            // Kick next tile's async global->LDS transfers (overlap with WMMAs).
#pragma unroll
            for (int j = 0; j < 8; ++j) {
                int i  = tid + j * 256;
                int r  = i >> 6;
                int c4 = i & 63;
                async_copy16(emb + ((size_t)(it + 1) * CODES_PER_ITER + r) * DIM + c4 * 4,
                             nxt + r * XPITCH + c4 * 4);
            }
            __builtin_amdgcn_s_wait_asynccnt(8);   // in-order: cur's 8 ops done
        } else {
            __builtin_amdgcn_s_wait_asynccnt(0);
        }
        __syncthreads();       // all waves' cur complete (and x-tile on it==0)
        const float* bbase = cur;
#else
        __syncthreads();       // previous tile fully consumed
#pragma unroll
        for (int j = 0; j < 8; ++j) {
            int i  = tid + j * 256;
            int r  = i >> 6;
            int c4 = i & 63;
            float* dst = ebuf0 + r * XPITCH + c4 * 4;
            dst[0] = pf[j].x; dst[1] = pf[j].y; dst[2] = pf[j].z; dst[3] = pf[j].w;
        }
        __syncthreads();
        const int nt = (it + 1 < NITER) ? it + 1 : it;
#pragma unroll
        for (int j = 0; j < 8; ++j) {
            int i  = tid + j * 256;
            int r  = i >> 6;
            int c4 = i & 63;
            pf[j] = *(const float4*)(emb +
                     ((size_t)nt * CODES_PER_ITER + r) * DIM + c4 * 4);
        }
        const float* bbase = ebuf0;
#endif

        // Two 16x16 dot tiles sharing the A fragment:
        // 128 x v_wmma_f32_16x16x4_f32, 3 ds_b64 loads per 2 WMMAs.
        const float* bptr0 = bbase + boff;
        const float* bptr1 = bptr0 + 16 * XPITCH;
        v8f acc0 = {}, acc1 = {};
#pragma unroll 8
        for (int d = 0; d < DIM; d += 4) {
            v2f a, b0, b1;
            a[0]  = aptr[d];   a[1]  = aptr[d + 1];
            b0[0] = bptr0[d];  b0[1] = bptr0[d + 1];
            b1[0] = bptr1[d];  b1[1] = bptr1[d + 1];
            acc0 = __builtin_amdgcn_wmma_f32_16x16x4_f32(
                false, a, false, b0, (short)0, acc0, false, false);
            acc1 = __builtin_amdgcn_wmma_f32_16x16x4_f32(
                false, a, false, b1, (short)0, acc1, false, false);
        }

        // score = ||e||^2 - 2*dot ; branchless lane-local running min.
        // Update tile0 before tile1: strict < keeps the earliest code (first-index).
        const float eq0 = lds_q[it * CODES_PER_ITER + l16];
        const float eq1 = lds_q[it * CODES_PER_ITER + 16 + l16];
#pragma unroll
        for (int r = 0; r < 8; ++r) {
            float v0 = fmaf(-2.f, acc0[r], eq0);
            bool t0 = v0 < best[r];
            best[r]  = t0 ? v0 : best[r];
            btile[r] = t0 ? (2 * it) : btile[r];

            float v1 = fmaf(-2.f, acc1[r], eq1);
            bool t1 = v1 < best[r];
            best[r]  = t1 ? v1 : best[r];
            btile[r] = t1 ? (2 * it + 1) : btile[r];
        }

#if HAVE_ASYNC_LDS
        __syncthreads();   // done reading cur before it is overwritten next iter
#endif
    }

    // One cross-lane argmin per register (16-lane halves), smaller-id tie-break.
#pragma unroll
    for (int r = 0; r < 8; ++r) {
        float val = best[r];
        int   id  = btile[r] * 16 + l16;
#pragma unroll
        for (int off = 1; off < 16; off <<= 1) {
            float ov = __shfl_xor(val, off, 32);
            int   oi = __shfl_xor(id, off, 32);
            bool take = (ov < val) || (ov == val && oi < id);
            val = take ? ov : val;
            id  = take ? oi : id;
        }
        btile[r] = id;   // final code index for row r + 8*half
    }

    if (l16 == 0) {
#pragma unroll
        for (int r = 0; r < 8; ++r)
            idx_out[rbase + wave * 16 + half * 8 + r] = btile[r];
    }
}

// ---------------- kernel 2: gather rows + per-block loss partials ----------------
__global__ void __launch_bounds__(256) vq_gather(const float* __restrict__ x,
                                                 const float* __restrict__ emb,
                                                 const int* __restrict__ idx,
                                                 float* __restrict__ out_q,
                                                 float* __restrict__ partial) {
    __shared__ int   sidx[64];
    __shared__ float sred[256];
    const int tid   = threadIdx.x;
    const int rbase = blockIdx.x * 64;

    if (tid < 64) sidx[tid] = idx[rbase + tid];
    __syncthreads();

    float s = 0.f;
    for (int r = 0; r < 64; ++r) {
        const int row = rbase + r;
        const int id  = sidx[r];
        float q  = emb[(size_t)id * DIM + tid];
        float xi = x[(size_t)row * DIM + tid];
        out_q[(size_t)row * DIM + tid] = q;   // straight-through == gathered rows
        float d = q - xi;
        s = fmaf(d, d, s);
    }

    sred[tid] = s;
    __syncthreads();
    for (int st = 128; st > 0; st >>= 1) {
        if (tid < st) sred[tid] += sred[tid + st];
        __syncthreads();
    }
    if (tid == 0) partial[blockIdx.x] = sred[0];
}

// ---------------- kernel 3: final deterministic reduction ----------------
__global__ void vq_loss(const float* __restrict__ partial, float* __restrict__ loss_out) {
    __shared__ float sred[256];
    const int tid = threadIdx.x;
    sred[tid] = partial[tid];
    __syncthreads();
    for (int st = 128; st > 0; st >>= 1) {
        if (tid < st) sred[tid] += sred[tid + st];
        __syncthreads();
    }
    // q_latent_loss + 0.25 * e_latent_loss, both == MSE(q, x)
    if (tid == 0) loss_out[0] = 1.25f * sred[0] / 4194304.0f;
}

extern "C" void kernel_launch(void* const* d_in, const int* in_sizes, int n_in,
                              void* d_out, int out_size, void* d_ws, size_t ws_size,
                              hipStream_t stream) {
    const float* x   = (const float*)d_in[0];   // [8,2048,256] f32
    const float* emb = (const float*)d_in[1];   // [1024,256]  f32
    float* out = (float*)d_out;                 // out[0]=loss, out[1..]=quantized

    // workspace layout: e_sq[1024] f32 | idx[16384] i32 | partial[256] f32
    float* esq     = (float*)d_ws;
    int*   idx     = (int*)((char*)d_ws + 1024 * sizeof(float));
    float* partial = (float*)((char*)d_ws + 1024 * sizeof(float) + 16384 * sizeof(int));

    vq_esq<<<EMB_K / 256, 256, 0, stream>>>(emb, esq);

    const size_t lds_bytes = (size_t)LDS_TOTAL_ELEMS * sizeof(float);
    vq_argmin<<<M_TOTAL / ROWS_PER_WG, 256, lds_bytes, stream>>>(x, emb, esq, idx);

    vq_gather<<<M_TOTAL / 64, 256, 0, stream>>>(x, emb, idx, out + 1, partial);
    vq_loss<<<1, 256, 0, stream>>>(partial, out);
}